// SparseMLP_66305705116130
// MI455X (gfx1250) — compile-verified
//
#include <hip/hip_runtime.h>
#include <math.h>

typedef float v2f __attribute__((ext_vector_type(2)));
typedef float v8f __attribute__((ext_vector_type(8)));
typedef unsigned int u32x4 __attribute__((ext_vector_type(4)));
typedef int i32x4 __attribute__((ext_vector_type(4)));
typedef int i32x8 __attribute__((ext_vector_type(8)));

#define ALPHA 1.702f
#define LIMIT 7.0f
#define TOPK  4

#define BS   2048      // B*S tokens
#define HH   1024
#define II   2048
#define EE   16

#define MT   16        // tokens per M-tile
#define KH   128       // H chunk staged in LDS for GEMM1
#define ICH  128       // I chunk per iteration (8 waves * 16 cols)
#define XPAD 4         // LDS row padding (dwords) to break bank conflicts

static __device__ inline v8f v8f_zero() {
    v8f z;
#pragma unroll
    for (int i = 0; i < 8; ++i) z[i] = 0.0f;
    return z;
}

// ---------------------------------------------------------------------------
// Router: one wave (32 lanes) per token. Lane e<16 owns expert e's logit.
// ---------------------------------------------------------------------------
__global__ __launch_bounds__(32) void moe_router_kernel(
    const float* __restrict__ hs,      // [BS, H]
    const float* __restrict__ rw,      // [E, H]
    const float* __restrict__ rb,      // [E]
    float* __restrict__ topkw_out,     // [BS, TOPK]
    int*   __restrict__ cnt,           // [E]
    int*   __restrict__ toklist,       // [E, BS]
    float* __restrict__ wlist)         // [E, BS]
{
    const int t    = blockIdx.x;
    const int lane = threadIdx.x;
    const float* x = hs + (size_t)t * HH;

    float logit = -1e30f;
    if (lane < EE) {
        const float* w = rw + (size_t)lane * HH;
        float acc = 0.0f;
        for (int h = 0; h < HH; ++h) acc += x[h] * w[h];
        logit = acc + rb[lane];
    }

    float m = logit;
#pragma unroll
    for (int off = 16; off; off >>= 1) m = fmaxf(m, __shfl_xor(m, off, 32));
    float p = (lane < EE) ? expf(logit - m) : 0.0f;
    float s = p;
#pragma unroll
    for (int off = 16; off; off >>= 1) s += __shfl_xor(s, off, 32);
    p = p / s;

    float myp = p;
    float tw[TOPK];
    int   te[TOPK];
    float wsum = 0.0f;
#pragma unroll
    for (int k = 0; k < TOPK; ++k) {
        float v = myp;
        int   idx = lane;
#pragma unroll
        for (int off = 16; off; off >>= 1) {
            float ov = __shfl_xor(v, off, 32);
            int   oi = __shfl_xor(idx, off, 32);
            if (ov > v || (ov == v && oi < idx)) { v = ov; idx = oi; }
        }
        tw[k] = v; te[k] = idx; wsum += v;
        if (lane == idx) myp = -1.0f;
    }

    if (lane < TOPK) {
        float wn = tw[lane] / wsum;
        topkw_out[(size_t)t * TOPK + lane] = wn;
        int ex  = te[lane];
        int pos = atomicAdd(&cnt[ex], 1);
        toklist[ex * BS + pos] = t;
        wlist[ex * BS + pos]   = wn;
    }
}

// ---------------------------------------------------------------------------
// Fused expert MLP: per (expert, 16-token tile). 8 waves.
// X tile gathered by the Tensor Data Mover (gather mode, 16x 16-bit row
// indices, 4-dword LDS pad per 128-dword row). GEMM1 f32 WMMA -> activation
// -> LDS -> GEMM2 f32 WMMA partials -> weighted atomic accumulate.
// ---------------------------------------------------------------------------
__global__ __launch_bounds__(256) void moe_expert_kernel(
    const float* __restrict__ hs,      // [BS, H]
    const float* __restrict__ w1,      // [E, H, 2I] interleaved gate/up cols
    const float* __restrict__ b1,      // [E, 2I]
    const float* __restrict__ w2,      // [E, I, H]
    const float* __restrict__ b2,      // [E, H]
    const int*   __restrict__ cnt,     // [E]
    const int*   __restrict__ toklist, // [E, BS]
    const float* __restrict__ wlist,   // [E, BS]
    float* __restrict__ out)           // [BS, H] (pre-zeroed)
{
    const int e    = blockIdx.x / (BS / MT);
    const int mt   = blockIdx.x % (BS / MT);
    const int ntok = cnt[e];
    if (mt * MT >= ntok) return;

    const int lane     = threadIdx.x & 31;
    const int wave     = threadIdx.x >> 5;    // 0..7
    const int row      = lane & 15;           // M row / N col within 16x16 block
    const int hi_half  = lane >> 4;           // 0 or 1

    __shared__ float lds_x[MT][KH + XPAD];
    __shared__ float lds_act[MT][ICH + XPAD];

    const int   gidx = mt * MT + row;
    const int   tok  = (gidx < ntok) ? toklist[e * BS + gidx] : 0;
    const float rw_  = (gidx < ntok) ? wlist[e * BS + gidx]   : 0.0f;

    const float* w1e = w1 + (size_t)e * HH * (2 * II);
    const float* w2e = w2 + (size_t)e * II * HH;

    // ---- TDM descriptor pieces that are h0-invariant ----
    // group1: wg_mask=0, data_size=2 (4B), pad_enable=1, pad_interval=6
    // (every 128 dwords), pad_amount=3 (4 dwords), tensor_dim0=H,
    // tensor_dim1=BS, tile_dim0=KH, tile_dim1=16 indices, dim0_stride=H.
    i32x8 tg1;
    tg1[0] = (2 << 16) | (1 << 20) | (6 << 22) | (3 << 25);
    tg1[1] = (int)((HH & 0xFFFFu) << 16);
    tg1[2] = (int)((BS & 0xFFFFu) << 16) | (int)((HH >> 16) & 0xFFFF);
    tg1[3] = (int)(KH << 16) | (int)((BS >> 16) & 0xFFFF);
    tg1[4] = MT;                  // tile_dim1 = #gather indices
    tg1[5] = HH;                  // tensor_dim0_stride (data_size units)
    tg1[6] = 0;
    tg1[7] = 0;

    // groups 2/3: 16x 16-bit gathered row (token) indices
    i32x4 tg2, tg3;
#pragma unroll
    for (int q = 0; q < 4; ++q) {
        int g0i = mt * MT + 2 * q;
        int g1i = g0i + 1;
        int tA = (g0i < ntok) ? toklist[e * BS + g0i] : 0;
        int tB = (g1i < ntok) ? toklist[e * BS + g1i] : 0;
        tg2[q] = (tA & 0xFFFF) | ((tB & 0xFFFF) << 16);
        int h0i = mt * MT + 8 + 2 * q;
        int h1i = h0i + 1;
        int tC = (h0i < ntok) ? toklist[e * BS + h0i] : 0;
        int tD = (h1i < ntok) ? toklist[e * BS + h1i] : 0;
        tg3[q] = (tC & 0xFFFF) | ((tD & 0xFFFF) << 16);
    }

    // extra descriptor group for the 6-arg toolchain form (unused -> zeros)
    i32x8 tg4;
#pragma unroll
    for (int q = 0; q < 8; ++q) tg4[q] = 0;

    const unsigned lds_x_base = (unsigned)(size_t)(&lds_x[0][0]); // low 32 = LDS offset

    v8f outacc[8];
#pragma unroll
    for (int nb = 0; nb < 8; ++nb) outacc[nb] = v8f_zero();

    for (int i0 = 0; i0 < II; i0 += ICH) {
        const int icol = i0 + wave * 16 + row;   // this lane's B-frag column (i index)
        v8f gate_acc = v8f_zero();
        v8f up_acc   = v8f_zero();

        for (int h0 = 0; h0 < HH; h0 += KH) {
            __syncthreads();   // previous readers of lds_x / lds_act done
            if (threadIdx.x < 32) {
                // group0: count=1, gather_mode=1 (16-bit idx), lds_addr,
                // 57-bit global addr of tile start (row 0, col h0), type=2
                unsigned long long ga =
                    (unsigned long long)(size_t)hs + (unsigned long long)h0 * 4ull;
                u32x4 tg0;
                tg0[0] = 0x80000001u;                 // count=1 | gather_mode
                tg0[1] = lds_x_base;
                tg0[2] = (unsigned)(ga & 0xFFFFFFFFull);
                tg0[3] = ((unsigned)(ga >> 32) & 0x01FFFFFFu) | 0x80000000u; // type=2
                __builtin_amdgcn_tensor_load_to_lds(tg0, tg1, tg2, tg3, tg4, 0);
                __builtin_amdgcn_s_wait_tensorcnt(0);
            }
            // prefetch next H-chunk of W1 one tile ahead
            if (h0 + KH < HH)
                __builtin_prefetch(&w1e[(size_t)(h0 + KH + 2 * hi_half) * (2 * II) + 2 * icol], 0, 0);
            __syncthreads();

#pragma unroll 4
            for (int k = 0; k < KH; k += 4) {
                const int kk = k + 2 * hi_half;
                v2f a;
                a.x = lds_x[row][kk + 0];
                a.y = lds_x[row][kk + 1];
                v2f bg, bu;
                const size_t r0 = (size_t)(h0 + kk) * (2 * II) + 2 * icol;
                const size_t r1 = r0 + (size_t)(2 * II);
                bg.x = w1e[r0];     bg.y = w1e[r1];
                bu.x = w1e[r0 + 1]; bu.y = w1e[r1 + 1];
                gate_acc = __builtin_amdgcn_wmma_f32_16x16x4_f32(
                    false, a, false, bg, (short)0, gate_acc, false, false);
                up_acc   = __builtin_amdgcn_wmma_f32_16x16x4_f32(
                    false, a, false, bu, (short)0, up_acc, false, false);
            }
        }

        // activation (column icol fixed per lane; rows m = r + 8*hi_half)
        const float bg_ = b1[(size_t)e * (2 * II) + 2 * icol];
        const float bu_ = b1[(size_t)e * (2 * II) + 2 * icol + 1];
#pragma unroll
        for (int r = 0; r < 8; ++r) {
            float g = gate_acc[r] + bg_;
            float u = up_acc[r]   + bu_;
            g = fminf(g, LIMIT);
            u = fminf(fmaxf(u, -LIMIT), LIMIT);
            const float glu = g / (1.0f + expf(-ALPHA * g));
            const float act = (u + 1.0f) * glu;
            lds_act[r + 8 * hi_half][wave * 16 + row] = act;
        }
        // prefetch next I-chunk of W2 one tile ahead
        if (i0 + ICH < II)
            __builtin_prefetch(&w2e[(size_t)(i0 + ICH + 2 * hi_half) * HH + wave * 128 + row], 0, 0);
        __syncthreads();

        // GEMM2 partial: out[m][wave*128 + nb*16 + n] += act[m][i0..] @ W2
#pragma unroll
        for (int nb = 0; nb < 8; ++nb) {
            const int h = wave * 128 + nb * 16 + row;
#pragma unroll 4
            for (int k = 0; k < ICH; k += 4) {
                const int kk = k + 2 * hi_half;
                v2f a;
                a.x = lds_act[row][kk + 0];
                a.y = lds_act[row][kk + 1];
                v2f b;
                b.x = w2e[(size_t)(i0 + kk + 0) * HH + h];
                b.y = w2e[(size_t)(i0 + kk + 1) * HH + h];
                outacc[nb] = __builtin_amdgcn_wmma_f32_16x16x4_f32(
                    false, a, false, b, (short)0, outacc[nb], false, false);
            }
        }
    }

    // epilogue: out[tok_m][h] += w_m * (acc + down_bias[h])
#pragma unroll
    for (int nb = 0; nb < 8; ++nb) {
        const int h = wave * 128 + nb * 16 + row;
        const float bias = b2[(size_t)e * HH + h];
#pragma unroll
        for (int r = 0; r < 8; ++r) {
            const int m = r + 8 * hi_half;
            const int   tok_m = __shfl(tok, m, 32);
            const float w_m   = __shfl(rw_, m, 32);
            atomicAdd(&out[(size_t)tok_m * HH + h], w_m * (outacc[nb][r] + bias));
        }
    }
}

// ---------------------------------------------------------------------------
// d_out = [ output (BS*H floats) | topk_weights (BS*4 floats) ]
// d_ws  = [ cnt (256B) | toklist E*BS ints | wlist E*BS floats ] ~= 256 KB
// ---------------------------------------------------------------------------
extern "C" void kernel_launch(void* const* d_in, const int* in_sizes, int n_in,
                              void* d_out, int out_size, void* d_ws, size_t ws_size,
                              hipStream_t stream) {
    const float* hs = (const float*)d_in[0];
    const float* rw = (const float*)d_in[1];
    const float* rb = (const float*)d_in[2];
    const float* w1 = (const float*)d_in[3];
    const float* b1 = (const float*)d_in[4];
    const float* w2 = (const float*)d_in[5];
    const float* b2 = (const float*)d_in[6];

    float* out   = (float*)d_out;
    float* topkw = out + (size_t)BS * HH;

    int*   cnt     = (int*)d_ws;
    int*   toklist = (int*)((char*)d_ws + 256);
    float* wlist   = (float*)((char*)d_ws + 256 + (size_t)EE * BS * sizeof(int));

    (void)hipMemsetAsync(d_out, 0, (size_t)BS * HH * sizeof(float), stream);
    (void)hipMemsetAsync(d_ws, 0, 256, stream);

    moe_router_kernel<<<BS, 32, 0, stream>>>(hs, rw, rb, topkw, cnt, toklist, wlist);
    moe_expert_kernel<<<EE * (BS / MT), 256, 0, stream>>>(
        hs, w1, b1, w2, b2, cnt, toklist, wlist, out);
}